// GraphemeAligner_29755533427070
// MI455X (gfx1250) — compile-verified
//
#include <hip/hip_runtime.h>
#include <cmath>

// Problem constants (from reference)
#define BATCH     64
#define T_FRAMES  4000
#define L_TOK     512
#define VOCAB     32
#define TS        200                  // frames per LDS tile
#define NT        (T_FRAMES / TS)      // 20 tiles
#define WORDS_PER_T  (T_FRAMES / 32)   // 125 packed decision words per column
#define BITS_PER_B   (WORDS_PER_T * L_TOK)  // 64000 words per batch
#define CPL       16                   // columns per lane (32 lanes * 16 = 512)

typedef unsigned int uint;
typedef uint  u32x4 __attribute__((ext_vector_type(4)));
typedef int   i32x4 __attribute__((ext_vector_type(4)));
typedef int   i32x8 __attribute__((ext_vector_type(8)));

__device__ __forceinline__ uint lds_off(const void* p) {
  // Generic pointer to a __shared__ object: low 32 bits are the LDS byte offset
  return (uint)(((unsigned long long)(uintptr_t)p) & 0xFFFFFFFFull);
}

// Issue a TDM load of a [ts x 32] f32 tile (row-major, contiguous) into LDS.
// D# per CDNA5 ISA ch.7/8: group0 = {count, lds_addr, global_addr, type=2},
// group1 = {data_size=4B, tensor dims/strides, tile dims}. 2D tile => groups 2/3 zero.
__device__ __forceinline__ void tdm_load_tile(const float* gsrc, uint ldsaddr, int ts) {
  unsigned long long ga = (unsigned long long)(uintptr_t)gsrc;
  u32x4 g0;
  g0.x = 1u;                                   // count=1, is_restore=0, gather off
  g0.y = ldsaddr;                              // lds_addr (bytes)
  g0.z = (uint)(ga & 0xFFFFFFFFu);             // global_addr[31:0]
  g0.w = (uint)((ga >> 32) & 0x01FFFFFFu)      // global_addr[56:32]
       | (2u << 30);                           // type=2 ("image")
  i32x8 g1;
  g1[0] = (int)(2u << 16);                     // workgroup_mask=0, data_size=2 (4 bytes)
  g1[1] = (int)((uint)VOCAB << 16);            // tensor_dim0=32 (low16 @ bits 63:48)
  g1[2] = (int)(((uint)ts & 0xFFFFu) << 16);   // tensor_dim0 hi=0 | tensor_dim1 low16
  g1[3] = (int)((uint)VOCAB << 16);            // tensor_dim1 hi=0 | tile_dim0=32
  g1[4] = ts;                                  // tile_dim1=ts | tile_dim2=0
  g1[5] = VOCAB;                               // tensor_dim0_stride = 32 elements
  g1[6] = 0;                                   // stride0 hi | stride1 lo (unused, 2D)
  g1[7] = 0;
  i32x4 z4 = {0, 0, 0, 0};
#if defined(__clang_major__) && (__clang_major__ >= 23)
  i32x8 z8 = {0, 0, 0, 0, 0, 0, 0, 0};
  __builtin_amdgcn_tensor_load_to_lds(g0, g1, z4, z4, z8, 0);
#else
  __builtin_amdgcn_tensor_load_to_lds(g0, g1, z4, z4, 0);
#endif
}

// Forward pass: ONE wave32 per batch element. Lane l owns trellis columns
// l*16+1 .. l*16+16 in registers (Jacobi recurrence: new[j] = f(old[j], old[j-1]),
// so only one __shfl_up per step, no barriers, no LDS row traffic).
// Emissions streamed HBM->LDS with the Tensor Data Mover (double buffered) and
// log-softmax-normalized IN PLACE by the parallel pre-pass, so the 4000-step
// sequential loop carries no per-column lse subtract and no lse load.
__global__ __launch_bounds__(32)
void ga_forward(const float* __restrict__ em, const int* __restrict__ tokens,
                unsigned* __restrict__ bits, int* __restrict__ tstart_out) {
  __shared__ float tile[2][TS * VOCAB];  // logits -> normalized log-probs in place

  const int b    = blockIdx.x;
  const int lane = threadIdx.x;

  const float* embase = em + (size_t)b * T_FRAMES * VOCAB;

  int tokv[CPL];
  #pragma unroll
  for (int i = 0; i < CPL; ++i)
    tokv[i] = tokens[b * L_TOK + lane * CPL + i];   // in [1, VOCAB)

  float    r[CPL];                 // trellis row (columns lane*16+1 .. +16)
  unsigned acc[CPL];               // per-column packed decision bits (32 steps)
  #pragma unroll
  for (int i = 0; i < CPL; ++i) { r[i] = -INFINITY; acc[i] = 0u; }

  tdm_load_tile(embase, lds_off(&tile[0][0]), TS);   // tile 0

  float best  = -INFINITY;         // running max of trellis[:, L] (lane 31 only used)
  int   bestT = 0;

  for (int k = 0; k < NT; ++k) {
    const int buf = k & 1;
    __builtin_amdgcn_s_wait_tensorcnt(0);            // tile k landed in LDS
    asm volatile("" ::: "memory");
    if ((k + 1) < NT) {
      // our older ds_loads (tile k-1) must drain before the TDM engine rewrites
      // that buffer; TDM is an independent agent so fence explicitly.
      asm volatile("s_wait_dscnt 0" ::: "memory");
      tdm_load_tile(embase + (size_t)(k + 1) * TS * VOCAB,
                    lds_off(&tile[(k + 1) & 1][0]), TS);
    }

    // Per-frame log-softmax, applied in place (frames are lane-private here).
    // Lane-rotated accesses stay conflict-free across the 64 LDS banks.
    for (int f = lane; f < TS; f += 32) {
      float* fr = &tile[buf][f * VOCAB];
      float m = -INFINITY;
      #pragma unroll
      for (int v = 0; v < VOCAB; ++v) m = fmaxf(m, fr[(v + lane) & (VOCAB - 1)]);
      float s = 0.0f;
      #pragma unroll
      for (int v = 0; v < VOCAB; ++v) s += __expf(fr[(v + lane) & (VOCAB - 1)] - m);
      const float lse = m + __logf(s);
      #pragma unroll
      for (int v = 0; v < VOCAB; ++v) {
        const int vv = (v + lane) & (VOCAB - 1);
        fr[vv] = fr[vv] - lse;
      }
    }
    asm volatile("" ::: "memory");   // DS ops are in-order; just stop reordering

    // Sequential recurrence: 200 steps, all state in registers, ~6 VALU/column.
    for (int f = 0; f < TS; ++f) {
      const int    s  = k * TS + f;                  // global step -> trellis[s+1]
      const float* fr = &tile[buf][f * VOCAB];       // normalized log-probs
      const float  e0 = fr[0];                       // em[t][blank]
      const unsigned m = 1u << (s & 31);

      float bnd = __shfl_up(r[CPL - 1], 1, 32);      // old row, neighbor's top column
      if (lane == 0) bnd = 0.0f;                     // trellis[:,0] == 0

      #pragma unroll
      for (int i = CPL - 1; i >= 1; --i) {           // descending: r[i-1] still "old"
        const float et   = fr[tokv[i]];
        const float stay = r[i]     + e0;
        const float adv  = r[i - 1] + et;
        acc[i] = (adv > stay) ? (acc[i] | m) : acc[i];
        r[i]   = fmaxf(stay, adv);
      }
      {
        const float et   = fr[tokv[0]];
        const float stay = r[0] + e0;
        const float adv  = bnd  + et;
        acc[0] = (adv > stay) ? (acc[0] | m) : acc[0];
        r[0]   = fmaxf(stay, adv);
      }

      // lane 31's r[15] is trellis[s+1][512]; branchless first-argmax tracking
      bestT = (r[CPL - 1] > best) ? (s + 1) : bestT;
      best  = fmaxf(best, r[CPL - 1]);

      if ((s & 31) == 31) {                          // flush 32 packed steps
        unsigned* dst = bits + (size_t)b * BITS_PER_B
                             + (size_t)(s >> 5) * L_TOK + lane * CPL;
        #pragma unroll
        for (int q = 0; q < CPL / 4; ++q) {          // 4x b128, coalesced across wave
          u32x4 w; w.x = acc[4*q]; w.y = acc[4*q+1]; w.z = acc[4*q+2]; w.w = acc[4*q+3];
          *(u32x4*)(dst + 4 * q) = w;
        }
        #pragma unroll
        for (int i = 0; i < CPL; ++i) acc[i] = 0u;
      }
    }
  }
  if (lane == 31) tstart_out[b] = bestT;
}

// Backtrack: one lane per batch walks t = t_start..1 choosing stay/advance from
// the packed decision bits (16 MB total -> L2-resident). Output pre-zeroed.
__global__ __launch_bounds__(512)
void ga_backtrack(const unsigned* __restrict__ bits, const int* __restrict__ tstart,
                  float* __restrict__ out) {
  const int b = blockIdx.x;
  out[b * L_TOK + threadIdx.x] = 0.0f;
  __syncthreads();
  if (threadIdx.x != 0) return;

  const unsigned* bb = bits + (size_t)b * BITS_PER_B;
  const float invT = 1.0f / (float)T_FRAMES;
  int   jcol = L_TOK;
  float cnt  = 0.0f;
  int   t    = tstart[b];
  while (t >= 1 && jcol >= 1) {
    cnt += 1.0f;
    const int s = t - 1;
    const unsigned w = bb[(size_t)(s >> 5) * L_TOK + (jcol - 1)];
    if ((w >> (s & 31)) & 1u) {          // advance: flush this token's duration
      out[b * L_TOK + (jcol - 1)] = cnt * invT;
      cnt = 0.0f;
      jcol -= 1;
    }
    --t;
  }
  if (jcol >= 1 && cnt > 0.0f)           // leftover frames at the last column
    out[b * L_TOK + (jcol - 1)] = cnt * invT;
}

extern "C" void kernel_launch(void* const* d_in, const int* in_sizes, int n_in,
                              void* d_out, int out_size, void* d_ws, size_t ws_size,
                              hipStream_t stream) {
  (void)in_sizes; (void)n_in; (void)out_size; (void)ws_size;
  const float* em     = (const float*)d_in[0];   // [64, 4000, 32] f32
  const int*   tokens = (const int*)d_in[1];     // [64, 512] i32
  unsigned* bits   = (unsigned*)d_ws;            // 64 * 64000 words = 16 MB
  int*      tstart = (int*)((char*)d_ws + (size_t)BATCH * BITS_PER_B * sizeof(unsigned));

  ga_forward  <<<BATCH, 32,    0, stream>>>(em, tokens, bits, tstart);
  ga_backtrack<<<BATCH, L_TOK, 0, stream>>>(bits, tstart, (float*)d_out);
}